// H2GCN_Net_15530601743024
// MI455X (gfx1250) — compile-verified
//
#include <hip/hip_runtime.h>
#include <hip/hip_bf16.h>

// H2GCN inference for MI455X (gfx1250, wave32).
// Workspace layout: rbuf[N][448]  = [ r0(64) | r1a(64) | r1b(64) | r2a(128) | r2b(128) ]
// cols:                              0..63    64..127   128..191  192..319   320..447

typedef float v2f __attribute__((ext_vector_type(2)));
typedef float v8f __attribute__((ext_vector_type(8)));

#define F_IN 256
#define HH   64
#define CC   16
#define RTOT 448   // (2^(K+1)-1)*H = 7*64

// ---------------------------------------------------------------------------
// Zero the workspace (grid-stride, float4 stores)
// ---------------------------------------------------------------------------
__global__ void __launch_bounds__(256)
zero_ws_kernel(float4* __restrict__ p, long long n4) {
    long long stride = (long long)gridDim.x * blockDim.x;
    for (long long i = (long long)blockIdx.x * blockDim.x + threadIdx.x; i < n4; i += stride) {
        p[i] = make_float4(0.f, 0.f, 0.f, 0.f);
    }
}

// ---------------------------------------------------------------------------
// r0 = relu(x @ w_embed)   x:[N,256] fp32, w:[256,64] fp32  -> rbuf cols 0..63
// One wave computes a 16x64 slab using V_WMMA_F32_16X16X4_F32.
// A 16x4 layout : lane = {m=lane&15 (row), kh=lane>>4}; VGPR0/1 = K = 2kh, 2kh+1
// B 4x16 layout : lane = {n=lane&15 (col), kh=lane>>4}; VGPR0/1 = K = 2kh, 2kh+1
// D 16x16 layout: VGPR j -> row j + 8*kh, col = lane&15
// ---------------------------------------------------------------------------
__global__ void __launch_bounds__(256)
embed_gemm_relu(const float* __restrict__ x, const float* __restrict__ w,
                float* __restrict__ rbuf, int N, int tilesM)
{
    int wid = blockIdx.x * (blockDim.x >> 5) + (threadIdx.x >> 5);
    if (wid >= tilesM) return;                 // wave-uniform: EXEC stays all-1s
    int lane = threadIdx.x & 31;
    int m    = lane & 15;
    int kh   = lane >> 4;                      // 0 or 1
    int row0 = wid * 16;
    int rowA = row0 + m; if (rowA >= N) rowA = N - 1;   // clamped load (N%16==0 normally)

    const float* xrow = x + (long long)rowA * F_IN + 2 * kh;

    v8f acc0 = {}, acc1 = {}, acc2 = {}, acc3 = {};
    for (int k = 0; k < F_IN; k += 4) {
        v2f a = *(const v2f*)(xrow + k);                      // K = k+2kh, k+2kh+1
        const float* wp = w + (long long)(k + 2 * kh) * HH + m;
        v2f b0, b1, b2, b3;
        b0.x = wp[0];      b0.y = wp[HH];
        b1.x = wp[16];     b1.y = wp[HH + 16];
        b2.x = wp[32];     b2.y = wp[HH + 32];
        b3.x = wp[48];     b3.y = wp[HH + 48];
        acc0 = __builtin_amdgcn_wmma_f32_16x16x4_f32(false, a, false, b0, (short)0, acc0, false, false);
        acc1 = __builtin_amdgcn_wmma_f32_16x16x4_f32(false, a, false, b1, (short)0, acc1, false, false);
        acc2 = __builtin_amdgcn_wmma_f32_16x16x4_f32(false, a, false, b2, (short)0, acc2, false, false);
        acc3 = __builtin_amdgcn_wmma_f32_16x16x4_f32(false, a, false, b3, (short)0, acc3, false, false);
    }

    #pragma unroll
    for (int j = 0; j < 8; ++j) {
        int row = row0 + j + 8 * kh;
        if (row < N) {
            float* o = rbuf + (long long)row * RTOT;
            o[ 0 + m] = fmaxf(acc0[j], 0.f);
            o[16 + m] = fmaxf(acc1[j], 0.f);
            o[32 + m] = fmaxf(acc2[j], 0.f);
            o[48 + m] = fmaxf(acc3[j], 0.f);
        }
    }
}

// ---------------------------------------------------------------------------
// COO SpMM (edge-parallel): out[row, outCol+f] += val * in[col, inCol+f]
// One wave per edge; lanes cover features (coalesced gather + f32 atomics,
// L2-resident since r-slices are 12.8-25.6MB << 192MB L2).
// Disjoint column ranges of rbuf => no read/write overlap within a launch.
// ---------------------------------------------------------------------------
__global__ void __launch_bounds__(256)
spmm_edges(const int* __restrict__ idx, const float* __restrict__ val, int E,
           float* rbuf, int inCol, int outCol, int Hin)
{
    int wid = blockIdx.x * (blockDim.x >> 5) + (threadIdx.x >> 5);
    if (wid >= E) return;
    int lane = threadIdx.x & 31;
    int row = idx[wid];          // idx[0][e]
    int col = idx[E + wid];      // idx[1][e]
    float v = val[wid];

    const float* src = rbuf + (long long)col * RTOT + inCol;
    float*       dst = rbuf + (long long)row * RTOT + outCol;

    // Prefetch next edge's gather row (global_prefetch_b8)
    if (wid + 1 < E) {
        int ncol = idx[E + wid + 1];
        __builtin_prefetch(rbuf + (long long)ncol * RTOT + inCol + lane, 0, 3);
    }

    for (int f = lane; f < Hin; f += 32) {
        atomicAdd(dst + f, v * src[f]);
    }
}

// ---------------------------------------------------------------------------
// In-place ReLU on rbuf[:, col0 : col0+ncols]
// ---------------------------------------------------------------------------
__global__ void __launch_bounds__(256)
relu_region(float* rbuf, int N, int col0, int ncols)
{
    long long t = (long long)blockIdx.x * blockDim.x + threadIdx.x;
    long long total = (long long)N * ncols;
    if (t >= total) return;
    long long r = t / ncols;
    int       c = (int)(t - r * ncols);
    float* p = rbuf + r * RTOT + col0 + c;
    float x = *p;
    *p = fmaxf(x, 0.f);
}

// ---------------------------------------------------------------------------
// out = softmax(rbuf @ w_classify)   rbuf:[N,448], wc:[448,16] -> out:[N,16]
// One wave per 16-row tile; K=448 in steps of 4 via V_WMMA_F32_16X16X4_F32.
// Softmax over the 16 output columns = reduction across a 16-lane group
// (xor-shuffle masks 1/2/4/8 never cross the 16-lane boundary on wave32).
// ---------------------------------------------------------------------------
__global__ void __launch_bounds__(256)
classify_softmax(const float* __restrict__ rbuf, const float* __restrict__ wc,
                 float* __restrict__ out, int N, int tilesM)
{
    int wid = blockIdx.x * (blockDim.x >> 5) + (threadIdx.x >> 5);
    if (wid >= tilesM) return;                 // wave-uniform
    int lane = threadIdx.x & 31;
    int m    = lane & 15;
    int kh   = lane >> 4;
    int row0 = wid * 16;
    int rowA = row0 + m; if (rowA >= N) rowA = N - 1;

    const float* rrow = rbuf + (long long)rowA * RTOT + 2 * kh;

    v8f acc = {};
    for (int k = 0; k < RTOT; k += 4) {
        v2f a = *(const v2f*)(rrow + k);
        const float* wp = wc + (long long)(k + 2 * kh) * CC + m;
        v2f b; b.x = wp[0]; b.y = wp[CC];
        acc = __builtin_amdgcn_wmma_f32_16x16x4_f32(false, a, false, b, (short)0, acc, false, false);
    }

    #pragma unroll
    for (int j = 0; j < 8; ++j) {
        float v = acc[j];
        float mx = v;
        mx = fmaxf(mx, __shfl_xor(mx, 1));
        mx = fmaxf(mx, __shfl_xor(mx, 2));
        mx = fmaxf(mx, __shfl_xor(mx, 4));
        mx = fmaxf(mx, __shfl_xor(mx, 8));
        float e = __expf(v - mx);
        float s = e;
        s += __shfl_xor(s, 1);
        s += __shfl_xor(s, 2);
        s += __shfl_xor(s, 4);
        s += __shfl_xor(s, 8);
        int row = row0 + j + 8 * kh;
        if (row < N) out[(long long)row * CC + m] = e / s;
    }
}

// ---------------------------------------------------------------------------
extern "C" void kernel_launch(void* const* d_in, const int* in_sizes, int n_in,
                              void* d_out, int out_size, void* d_ws, size_t ws_size,
                              hipStream_t stream)
{
    const float* x      = (const float*)d_in[0];
    const int*   a1_idx = (const int*)  d_in[1];
    const float* a1_val = (const float*)d_in[2];
    const int*   a2_idx = (const int*)  d_in[3];
    const float* a2_val = (const float*)d_in[4];
    const float* w_emb  = (const float*)d_in[5];
    const float* w_cls  = (const float*)d_in[6];
    float* out  = (float*)d_out;
    float* rbuf = (float*)d_ws;

    const int N  = in_sizes[0] / F_IN;   // 50000
    const int E1 = in_sizes[2];          // 800000
    const int E2 = in_sizes[4];          // 1600000
    const int tilesM = (N + 15) / 16;    // 3125

    // 1) zero workspace (atomic accumulation targets)
    long long n4 = (long long)N * RTOT / 4;
    zero_ws_kernel<<<2048, 256, 0, stream>>>((float4*)rbuf, n4);

    // 2) r0 = relu(x @ w_embed) -> cols 0..63
    embed_gemm_relu<<<(tilesM + 7) / 8, 256, 0, stream>>>(x, w_emb, rbuf, N, tilesM);

    // 3) hop 1: read r0 (cols 0..63), write cols 64..191
    spmm_edges<<<(E1 + 7) / 8, 256, 0, stream>>>(a1_idx, a1_val, E1, rbuf, 0, 64,  HH);
    spmm_edges<<<(E2 + 7) / 8, 256, 0, stream>>>(a2_idx, a2_val, E2, rbuf, 0, 128, HH);
    {
        long long tot = (long long)N * 128;
        relu_region<<<(int)((tot + 255) / 256), 256, 0, stream>>>(rbuf, N, 64, 128);
    }

    // 4) hop 2: read r1 (cols 64..191, 128-wide), write cols 192..447
    spmm_edges<<<(E1 + 7) / 8, 256, 0, stream>>>(a1_idx, a1_val, E1, rbuf, 64, 192, 128);
    spmm_edges<<<(E2 + 7) / 8, 256, 0, stream>>>(a2_idx, a2_val, E2, rbuf, 64, 320, 128);
    {
        long long tot = (long long)N * 256;
        relu_region<<<(int)((tot + 255) / 256), 256, 0, stream>>>(rbuf, N, 192, 256);
    }

    // 5) logits = rbuf @ w_classify ; softmax -> out
    classify_softmax<<<(tilesM + 7) / 8, 256, 0, stream>>>(rbuf, w_cls, out, N, tilesM);
}